// CHSLoss2_81801947120186
// MI455X (gfx1250) — compile-verified
//
#include <hip/hip_runtime.h>
#include <hip/hip_bf16.h>

// Problem constants (fixed by the reference's setup_inputs)
#define B_      32
#define H_      192
#define W_      192
#define SZ_     8
#define HW_     (H_ * W_)          // 36864
#define HIN_    (H_ * SZ_)         // 1536
#define WIN_    (W_ * SZ_)         // 1536
#define RROWS_  (HIN_)             // 1536 rows after W-reduce
#define RCOLS_  (W_)               // 192 cols after W-reduce

typedef __attribute__((ext_vector_type(2))) float v2f;
typedef __attribute__((ext_vector_type(8))) float v8f;

// ---------------------------------------------------------------------------
// Kernel 1: W-direction reduce. R[b][r][wo] = sum_{k<8} G[b][r][8*wo+k]
// Reads 302 MB (HBM-bound), writes 37.7 MB (L2-resident afterwards).
// Each thread: 2x float4 loads (32B contiguous), 1 store. Fully coalesced.
// ---------------------------------------------------------------------------
__global__ void pool_w_kernel(const float* __restrict__ G, float* __restrict__ R) {
    const long long P = (long long)B_ * RROWS_ * RCOLS_;   // 9,437,184
    long long i = (long long)blockIdx.x * blockDim.x + threadIdx.x;
    if (i >= P) return;
    int col  = (int)(i % RCOLS_);
    long long rowb = i / RCOLS_;                 // = batch*1536 + row
    const float4* src = (const float4*)(G + rowb * (long long)WIN_ + (long long)col * SZ_);
    float4 x = src[0];
    float4 y = src[1];
    R[i] = ((x.x + x.y) + (x.z + x.w)) + ((y.x + y.y) + (y.z + y.w));
}

// ---------------------------------------------------------------------------
// Kernel 2: H-direction reduce via V_WMMA_F32_16X16X4_F32.
// dmap 16x16 tile = sum_{kc=0..31} A(kc)[16x4] x B(kc)[4x16]
//   A(kc)[m][k] = (m == kc>>1) ? 1 : 0   (selector: rows 8m..8m+7 -> out row m)
//   B(kc)[k][n] = R[tileRow*128 + kc*4 + k][tileCol*16 + n]
// One wave per tile (EXEC all ones, wave32). 4608 tiles, 32 WMMA each.
// ---------------------------------------------------------------------------
__global__ void pool_h_wmma_kernel(const float* __restrict__ R, float* __restrict__ dmap) {
    int blk   = blockIdx.x;               // 0 .. 32*12*12-1
    int batch = blk / 144;
    int t     = blk % 144;
    int tr    = t / 12;                   // tile row (of 16 output rows)
    int tc    = t % 12;                   // tile col (of 16 output cols)
    int lane  = threadIdx.x;              // 0..31
    int half  = lane >> 4;                // 0: lanes 0-15, 1: lanes 16-31
    int l     = lane & 15;

    const float* Rb = R + (long long)batch * RROWS_ * RCOLS_;
    int col = tc * 16 + l;

    v8f acc = {};
    for (int kc = 0; kc < 32; ++kc) {
        // A: lanes 0-15 hold (M=l, K=0..1) in v0,v1; lanes 16-31 hold (M=l, K=2..3).
        // For this selector, A[m][k] is independent of k within the chunk.
        float aval = (l == (kc >> 1)) ? 1.0f : 0.0f;
        v2f a; a.x = aval; a.y = aval;
        // B: v0 = row K=0 (lanes 0-15) / K=2 (lanes 16-31); v1 = K=1 / K=3.
        int r0 = tr * 128 + kc * 4 + half * 2;
        v2f bv;
        bv.x = Rb[(long long)r0 * RCOLS_ + col];
        bv.y = Rb[(long long)(r0 + 1) * RCOLS_ + col];
        acc = __builtin_amdgcn_wmma_f32_16x16x4_f32(
            /*neg_a=*/false, a, /*neg_b=*/false, bv,
            /*c_mod=*/(short)0, acc, /*reuse_a=*/false, /*reuse_b=*/false);
    }

    // D layout: VGPR r -> (M=r, N=lane) for lanes 0-15; (M=r+8, N=lane-16) for 16-31.
    float* Db = dmap + (long long)batch * HW_;
    int outcol = tc * 16 + l;
    #pragma unroll
    for (int r = 0; r < 8; ++r) {
        int outrow = tr * 16 + r + half * 8;
        Db[(long long)outrow * W_ + outcol] = acc[r];
    }
}

// ---------------------------------------------------------------------------
// Kernel 3: exact k-th largest of err[i][b][:] via 4-pass radix select on the
// float bit pattern (valid: err >= 0 -> uint order == float order).
// One block per (i, b): 96 blocks. All data L2-resident.
// ---------------------------------------------------------------------------
__global__ void select_kernel(const float* __restrict__ dmap,
                              const float* __restrict__ o0,
                              const float* __restrict__ o1,
                              const float* __restrict__ o2,
                              const float* __restrict__ proc,
                              unsigned* __restrict__ thresh) {
    int i = blockIdx.x / B_;
    int b = blockIdx.x % B_;
    const float* ob = (i == 0 ? o0 : (i == 1 ? o1 : o2)) + (long long)b * HW_;
    const float* dm = dmap + (long long)b * HW_;

    int k0 = (int)((double)HW_ * 0.1 * (double)proc[0]);
    if (k0 < 1) {
        if (threadIdx.x == 0) thresh[blockIdx.x] = 0u;
        return;
    }

    __shared__ unsigned hist[256];
    __shared__ unsigned s_prefix;
    __shared__ int s_k;

    unsigned prefix = 0u, mask = 0u;
    int k = k0;

    for (int pass = 3; pass >= 0; --pass) {
        for (int t = threadIdx.x; t < 256; t += blockDim.x) hist[t] = 0u;
        __syncthreads();
        for (int e = threadIdx.x; e < HW_; e += blockDim.x) {
            float err = fabsf(dm[e] - ob[e]);
            unsigned bits = __float_as_uint(err);
            if ((bits & mask) == prefix)
                atomicAdd(&hist[(bits >> (pass * 8)) & 255u], 1u);
        }
        __syncthreads();
        if (threadIdx.x == 0) {
            unsigned cum = 0u;
            int sel = 0;
            for (int v = 255; v >= 0; --v) {
                if (cum + hist[v] >= (unsigned)k) { sel = v; break; }
                cum += hist[v];
            }
            s_prefix = prefix | ((unsigned)sel << (pass * 8));
            s_k = k - (int)cum;
        }
        __syncthreads();
        prefix = s_prefix;
        k = s_k;
        mask |= 0xFFu << (pass * 8);
        __syncthreads();
    }
    if (threadIdx.x == 0) thresh[blockIdx.x] = prefix;  // exact bits of v_min
}

// ---------------------------------------------------------------------------
// Kernel 4: masked squared-error accumulation over the 3 pairs (0,1),(0,2),(1,2)
// (or the num<1 fallback). Deterministic per-block partials.
// ---------------------------------------------------------------------------
__global__ void loss_kernel(const float* __restrict__ dmap,
                            const float* __restrict__ o0,
                            const float* __restrict__ o1,
                            const float* __restrict__ o2,
                            const float* __restrict__ proc,
                            const unsigned* __restrict__ thresh,
                            float* __restrict__ partials) {
    const int N = B_ * HW_;
    float wgt = 1.0f * proc[0];               // MAX_WEIGHT_RATIO * process
    int k0 = (int)((double)HW_ * 0.1 * (double)proc[0]);

    float acc = 0.0f;
    for (int idx = blockIdx.x * blockDim.x + threadIdx.x; idx < N;
         idx += gridDim.x * blockDim.x) {
        int b = idx / HW_;
        float d  = dmap[idx];
        float a0 = o0[idx], a1 = o1[idx], a2 = o2[idx];
        if (k0 < 1) {
            float e0 = a0 - d, e1 = a1 - d, e2 = a2 - d;
            acc += e0 * e0 + e1 * e1 + e2 * e2;
        } else {
            float t0 = __uint_as_float(thresh[0 * B_ + b]);
            float t1 = __uint_as_float(thresh[1 * B_ + b]);
            bool m0 = fabsf(d - a0) >= t0;
            bool m1 = fabsf(d - a1) >= t1;
            float c1 = wgt * a1 + (1.0f - wgt) * d;
            float c2 = wgt * a2 + (1.0f - wgt) * d;
            float s01 = m0 ? c1 : d;
            float s02 = m0 ? c2 : d;
            float s12 = m1 ? c2 : d;
            float e;
            e = a0 - s01; acc += e * e;
            e = a0 - s02; acc += e * e;
            e = a1 - s12; acc += e * e;
        }
    }

    __shared__ float red[256];
    red[threadIdx.x] = acc;
    __syncthreads();
    for (int s = 128; s > 0; s >>= 1) {
        if (threadIdx.x < (unsigned)s) red[threadIdx.x] += red[threadIdx.x + s];
        __syncthreads();
    }
    if (threadIdx.x == 0) partials[blockIdx.x] = red[0];
}

__global__ void finalize_kernel(const float* __restrict__ partials, int n,
                                float* __restrict__ out) {
    __shared__ float red[256];
    float a = 0.0f;
    for (int i = threadIdx.x; i < n; i += 256) a += partials[i];
    red[threadIdx.x] = a;
    __syncthreads();
    for (int s = 128; s > 0; s >>= 1) {
        if (threadIdx.x < (unsigned)s) red[threadIdx.x] += red[threadIdx.x + s];
        __syncthreads();
    }
    if (threadIdx.x == 0) out[0] = red[0];
}

// ---------------------------------------------------------------------------
extern "C" void kernel_launch(void* const* d_in, const int* in_sizes, int n_in,
                              void* d_out, int out_size, void* d_ws, size_t ws_size,
                              hipStream_t stream) {
    const float* out0 = (const float*)d_in[0];
    const float* out1 = (const float*)d_in[1];
    const float* out2 = (const float*)d_in[2];
    const float* gt   = (const float*)d_in[3];
    const float* proc = (const float*)d_in[4];

    // Workspace layout (floats): R | dmap | thresh(128) | partials(1024)
    float*    R        = (float*)d_ws;                              // 9,437,184
    float*    dmap     = R + (long long)B_ * RROWS_ * RCOLS_;       // 1,179,648
    unsigned* thresh   = (unsigned*)(dmap + (long long)B_ * HW_);   // 96 (pad 128)
    float*    partials = (float*)(thresh + 128);                    // 1024

    // 1) W-reduce: 9,437,184 outputs, 1/thread.
    {
        long long P = (long long)B_ * RROWS_ * RCOLS_;
        int blocks = (int)((P + 255) / 256);
        pool_w_kernel<<<blocks, 256, 0, stream>>>(gt, R);
    }
    // 2) H-reduce via WMMA: one wave per 16x16 dmap tile.
    pool_h_wmma_kernel<<<B_ * 12 * 12, 32, 0, stream>>>(R, dmap);
    // 3) Exact v_min per (i, b) via radix select.
    select_kernel<<<3 * B_, 256, 0, stream>>>(dmap, out0, out1, out2, proc, thresh);
    // 4) Pairwise masked squared-error partial sums.
    loss_kernel<<<1024, 256, 0, stream>>>(dmap, out0, out1, out2, proc, thresh, partials);
    // 5) Scalar finalize.
    finalize_kernel<<<1, 256, 0, stream>>>(partials, 1024, (float*)d_out);
}